// NERTagger_38835094290829
// MI455X (gfx1250) — compile-verified
//
#include <hip/hip_runtime.h>
#include <hip/hip_bf16.h>
#include <stddef.h>

// ---------------------------------------------------------------------------
// NER tagger: segment-sum words from encoder tokens, then words @ W_cls + b.
// B=8, S=4096, D=768, NT=9, W=B*S/2=16384.
// HBM-bound: 100.7 MB streamed once -> ~4.4us floor at 23.3 TB/s.
// GEMM tail done with V_WMMA_F32_16X16X4_F32 (full fp32 precision).
// ---------------------------------------------------------------------------

typedef __attribute__((ext_vector_type(2))) float v2f;
typedef __attribute__((ext_vector_type(8))) float v8f;

namespace ner {
constexpr int B  = 8;
constexpr int S  = 4096;
constexpr int D  = 768;
constexpr int NT = 9;
constexpr int W  = B * (S / 2);      // 16384 output words
constexpr int SCAN_T = 1024;         // threads in scan block
constexpr int TPT    = S / SCAN_T;   // tokens per thread in scan (4)
}

// --- Kernel 1: per-row scan of word starts (marker==2) ----------------------
// Writes wordStart[b*S + local_word_id] = token index; counts[b] = #words.
__global__ __launch_bounds__(ner::SCAN_T)
void ner_scan_kernel(const int* __restrict__ src, int* __restrict__ wordStart,
                     int* __restrict__ counts) {
  using namespace ner;
  const int b   = blockIdx.x;
  const int tid = threadIdx.x;
  const int* row = src + (size_t)b * S;

  int flags[TPT];
  int cnt = 0;
  const int base = tid * TPT;
#pragma unroll
  for (int i = 0; i < TPT; ++i) {
    const int m = row[base + i];
    flags[i] = (m == 2) ? 1 : 0;
    cnt += flags[i];
  }

  __shared__ int sc[SCAN_T];
  sc[tid] = cnt;
  __syncthreads();
  // Hillis-Steele inclusive scan over thread counts.
  for (int off = 1; off < SCAN_T; off <<= 1) {
    int v = (tid >= off) ? sc[tid - off] : 0;
    __syncthreads();
    sc[tid] += v;
    __syncthreads();
  }
  int excl = sc[tid] - cnt;   // exclusive prefix = first word id for this thread
#pragma unroll
  for (int i = 0; i < TPT; ++i) {
    if (flags[i]) {
      wordStart[(size_t)b * S + excl] = base + i;
      ++excl;
    }
  }
  if (tid == SCAN_T - 1) counts[b] = sc[SCAN_T - 1];
}

// --- Kernel 2: offsets[b] = prefix-sum of counts ----------------------------
__global__ void ner_offsets_kernel(const int* __restrict__ counts,
                                   int* __restrict__ offsets) {
  using namespace ner;
  if (threadIdx.x == 0 && blockIdx.x == 0) {
    int acc = 0;
    for (int b = 0; b < B; ++b) { offsets[b] = acc; acc += counts[b]; }
    offsets[B] = acc;
  }
}

// --- Kernel 3: fused word-sum + (words @ W_cls + b_cls) via f32 WMMA --------
// One block = 16 words (one WMMA M-tile). 256 threads = 8 waves.
__global__ __launch_bounds__(256)
void ner_fused_kernel(const float* __restrict__ enc,
                      const float* __restrict__ Wc,
                      const float* __restrict__ bias,
                      const int*  __restrict__ src,
                      const int*  __restrict__ wordStart,
                      const int*  __restrict__ counts,
                      const int*  __restrict__ offsets,
                      float* __restrict__ out) {
  using namespace ner;
  __shared__ float tileA[16 * D];        // 16 word-sum rows          (48 KB)
  __shared__ float Wlds[D * 16];         // W_cls zero-padded to N=16 (48 KB)
  __shared__ float partials[8 * 16 * 16];// per-wave C matrices       ( 8 KB)

  const int tile = blockIdx.x;
  const int tid  = threadIdx.x;

  // Stage W_cls [D][NT] -> Wlds [D][16], cols NT..15 zeroed.
  for (int i = tid; i < D * 16; i += 256) {
    const int kk = i >> 4, n = i & 15;
    Wlds[i] = (n < NT) ? Wc[(size_t)kk * NT + n] : 0.0f;
  }

  // ---- word sums: 16 groups x 16 threads; group g owns word tile*16+g ----
  const int g = tid >> 4;        // word within tile
  const int c = tid & 15;        // float4 column sub-lane
  const int w = tile * 16 + g;
  const int total = offsets[B];

  float4 acc[12];
#pragma unroll
  for (int i = 0; i < 12; ++i) acc[i] = make_float4(0.f, 0.f, 0.f, 0.f);

  if (w < total) {
    // locate row b: offsets[b] <= w < offsets[b+1]
    int b = 0;
    while (b < B - 1 && w >= offsets[b + 1]) ++b;
    const int lid = w - offsets[b];
    const int t0  = wordStart[(size_t)b * S + lid];
    const int t1  = (lid + 1 < counts[b]) ? wordStart[(size_t)b * S + lid + 1] : S;
    for (int tok = t0; tok < t1; ++tok) {
      const int m = src[(size_t)b * S + tok];
      if (m == 2 || m == 3) {
        const float4* p = (const float4*)(enc + (size_t)(b * S + tok) * D);
#pragma unroll
        for (int i = 0; i < 12; ++i) acc[i] += p[c + i * 16];
      }
    }
    // Last word of a row absorbs leading marker-3 tokens of the next row
    // (reference maps them to offsets[b+1]-1 via seg_in_row == -1).
    if (lid == counts[b] - 1 && b + 1 < B) {
      const int nb = b + 1;
      const int fs = (counts[nb] > 0) ? wordStart[(size_t)nb * S] : S;
      for (int tok = 0; tok < fs; ++tok) {
        if (src[(size_t)nb * S + tok] == 3) {
          const float4* p = (const float4*)(enc + (size_t)(nb * S + tok) * D);
#pragma unroll
          for (int i = 0; i < 12; ++i) acc[i] += p[c + i * 16];
        }
      }
    }
  }
#pragma unroll
  for (int i = 0; i < 12; ++i)
    *(float4*)&tileA[g * D + (c + i * 16) * 4] = acc[i];
  __syncthreads();

  // ---- GEMM: 8 waves each cover K-slice of 96, 24 x wmma_f32_16x16x4_f32 --
  const int wave = tid >> 5;
  const int lane = tid & 31;
  const int hi   = lane >> 4;      // 0: lanes 0-15 (K 0,1); 1: lanes 16-31 (K 2,3)
  const int mr   = lane & 15;      // A row (M) / B col (N)

  v8f cacc = {};
  const int kbeg = wave * 96;
#pragma unroll
  for (int k = kbeg; k < kbeg + 96; k += 4) {
    const int kk = k + (hi ? 2 : 0);
    // A fragment (16x4 f32, ISA 7.12.2): v0 = A[m][kk], v1 = A[m][kk+1]
    v2f a;
    a.x = tileA[mr * D + kk];
    a.y = tileA[mr * D + kk + 1];
    // B fragment (4x16 f32): v0 = B[kk][n], v1 = B[kk+1][n]
    v2f bb;
    bb.x = Wlds[kk * 16 + mr];
    bb.y = Wlds[(kk + 1) * 16 + mr];
    cacc = __builtin_amdgcn_wmma_f32_16x16x4_f32(
        false, a, false, bb, (short)0, cacc, false, false);
  }
  // C layout: VGPR r -> M = r + 8*hi, N = lane&15
#pragma unroll
  for (int r = 0; r < 8; ++r)
    partials[wave * 256 + (r + hi * 8) * 16 + mr] = cacc[r];
  __syncthreads();

  // reduce the 8 wave-partials, add bias, write [16 x NT]
  if (tid < 16 * NT) {
    const int m = tid / NT, n = tid % NT;
    float s = bias[n];
#pragma unroll
    for (int v = 0; v < 8; ++v) s += partials[v * 256 + m * 16 + n];
    out[(size_t)(tile * 16 + m) * NT + n] = s;
  }
}

extern "C" void kernel_launch(void* const* d_in, const int* in_sizes, int n_in,
                              void* d_out, int out_size, void* d_ws, size_t ws_size,
                              hipStream_t stream) {
  using namespace ner;
  (void)in_sizes; (void)n_in; (void)out_size; (void)ws_size;
  const float* enc  = (const float*)d_in[0];   // [B,S,D]
  const float* Wc   = (const float*)d_in[1];   // [D,NT]
  const float* bias = (const float*)d_in[2];   // [NT]
  const int*   src  = (const int*)d_in[3];     // [B,S]
  float* out = (float*)d_out;                  // [W,NT]

  int* wordStart = (int*)d_ws;                 // B*S ints (131072 B)
  int* counts    = wordStart + (size_t)B * S;  // B ints
  int* offsets   = counts + B;                 // B+1 ints

  ner_scan_kernel<<<B, SCAN_T, 0, stream>>>(src, wordStart, counts);
  ner_offsets_kernel<<<1, 32, 0, stream>>>(counts, offsets);
  ner_fused_kernel<<<W / 16, 256, 0, stream>>>(enc, Wc, bias, src, wordStart,
                                               counts, offsets, out);
}